// Model_1932735283607
// MI455X (gfx1250) — compile-verified
//
#include <hip/hip_runtime.h>

// RNN: h' = tanh(x @ W_ih^T + b_ih + h @ W_hh^T + b_hh), 2048 steps, then h @ W_out^T + b_out.
// One workgroup per 16-row batch tile (32 WGs, 8 wave32 waves each; one 16x16 N-tile per wave).
// All weight (B) fragments live in VGPRs for the whole sequence; x A-fragments are
// software-pipelined one step ahead; h ping-pongs through a small LDS buffer.

#define SEQ    2048
#define BATCH  512
#define INPT   64
#define HIDDEN 128
#define MTILE  16
#define HSTR   132   // padded h row stride (floats) -> conflict-free ds_load_b64 A fetches

typedef __attribute__((ext_vector_type(2))) float v2f;
typedef __attribute__((ext_vector_type(8))) float v8f;

__global__ __launch_bounds__(256, 1) void rnn_fused_kernel(
    const float* __restrict__ xs,     // [SEQ][BATCH][INPT]
    const float* __restrict__ W_ih,   // [HIDDEN][INPT]
    const float* __restrict__ W_hh,   // [HIDDEN][HIDDEN]
    const float* __restrict__ b_ih,   // [HIDDEN]
    const float* __restrict__ b_hh,   // [HIDDEN]
    const float* __restrict__ W_out,  // [1][HIDDEN]
    const float* __restrict__ b_out,  // [1]
    float* __restrict__ out)          // [BATCH][1]
{
    __shared__ float hbuf[2 * MTILE * HSTR];   // 16.5 KB ping-pong h tile

    const int tid  = threadIdx.x;
    const int wave = tid >> 5;                 // 8 waves; wave owns columns [N0, N0+16)
    const int lane = tid & 31;
    const int m    = lane & 15;                // A-matrix row (M)
    const int lh   = lane >> 4;                // lane half -> K sub-pair select
    const int nn   = lane & 15;                // B/C/D column within the tile
    const int b0   = blockIdx.x * MTILE;       // batch tile base
    const int N0   = wave * 16;

    // h0 = 0
    for (int i = tid; i < 2 * MTILE * HSTR; i += 256) hbuf[i] = 0.0f;

    const float bias = b_ih[N0 + nn] + b_hh[N0 + nn];

    // ---- loop-invariant B fragments -> registers (one-time, contiguous b64 reads) ----
    // B[k][n] = W[n][k]; vgpr v of chunk kc holds k = 4*kc + 2*lh + v, col n = N0+nn.
    v2f Bih[INPT / 4], Bhh[HIDDEN / 4];
#pragma unroll
    for (int kc = 0; kc < INPT / 4; ++kc)
        Bih[kc] = *(const v2f*)(W_ih + (size_t)(N0 + nn) * INPT + 4 * kc + 2 * lh);
#pragma unroll
    for (int kc = 0; kc < HIDDEN / 4; ++kc)
        Bhh[kc] = *(const v2f*)(W_hh + (size_t)(N0 + nn) * HIDDEN + 4 * kc + 2 * lh);

    // ---- preload x A-fragments for t = 0 ----
    v2f xa[INPT / 4];
    {
        const float* x0 = xs + (size_t)(b0 + m) * INPT;
#pragma unroll
        for (int kc = 0; kc < INPT / 4; ++kc)
            xa[kc] = *(const v2f*)(x0 + 4 * kc + 2 * lh);
    }
    __syncthreads();

    for (int t = 0; t < SEQ; ++t) {
        const float* hsrc = hbuf + (t & 1) * (MTILE * HSTR);
        float*       hdst = hbuf + ((t + 1) & 1) * (MTILE * HSTR);

        // batch all h A-fragment LDS loads up front (one dscnt wait, not 32)
        v2f ha[HIDDEN / 4];
#pragma unroll
        for (int kc = 0; kc < HIDDEN / 4; ++kc)
            ha[kc] = *(const v2f*)(hsrc + m * HSTR + 4 * kc + 2 * lh);

        v8f acc;
#pragma unroll
        for (int v = 0; v < 8; ++v) acc[v] = bias;

        // ---- x_t @ W_ih^T : K = 64 (A + B both already in registers) ----
#pragma unroll
        for (int kc = 0; kc < INPT / 4; ++kc)
            acc = __builtin_amdgcn_wmma_f32_16x16x4_f32(false, xa[kc], false, Bih[kc],
                                                        (short)0, acc, false, false);

        // ---- refill xa for t+1; completes under the hh GEMM + tanh + barrier ----
        {
            const int tn = (t + 1 < SEQ) ? (t + 1) : t;
            const int tp = (t + 2 < SEQ) ? (t + 2) : tn;
            const float* xnext = xs + ((size_t)tn * BATCH + b0 + m) * INPT;
            __builtin_prefetch(xs + ((size_t)tp * BATCH + b0 + m) * INPT, 0, 0);
#pragma unroll
            for (int kc = 0; kc < INPT / 4; ++kc)
                xa[kc] = *(const v2f*)(xnext + 4 * kc + 2 * lh);
        }

        // ---- h_t @ W_hh^T : K = 128 ----
#pragma unroll
        for (int kc = 0; kc < HIDDEN / 4; ++kc)
            acc = __builtin_amdgcn_wmma_f32_16x16x4_f32(false, ha[kc], false, Bhh[kc],
                                                        (short)0, acc, false, false);

        // ---- tanh + write h_{t+1} (C/D layout: row = v + 8*lh, col = nn) ----
#pragma unroll
        for (int v = 0; v < 8; ++v) {
            // tanh(x) = 1 - 2/(exp2(2*log2e*x)+1): exact saturation at +/-inf
            const float e = __builtin_amdgcn_exp2f(acc[v] * 2.885390081777927f);
            hdst[(v + 8 * lh) * HSTR + N0 + nn] = 1.0f - 2.0f / (e + 1.0f);
        }
        __syncthreads();
    }

    // ---- output projection: final h is in buffer 0 (SEQ is even) ----
    if (tid < MTILE) {
        float s = b_out[0];
#pragma unroll 8
        for (int n = 0; n < HIDDEN; ++n) s += hbuf[tid * HSTR + n] * W_out[n];
        out[b0 + tid] = s;
    }
}

extern "C" void kernel_launch(void* const* d_in, const int* in_sizes, int n_in,
                              void* d_out, int out_size, void* d_ws, size_t ws_size,
                              hipStream_t stream) {
    const float* xs    = (const float*)d_in[0];
    const float* W_ih  = (const float*)d_in[1];
    const float* W_hh  = (const float*)d_in[2];
    const float* b_ih  = (const float*)d_in[3];
    const float* b_hh  = (const float*)d_in[4];
    const float* W_out = (const float*)d_in[5];
    const float* b_out = (const float*)d_in[6];
    float* out = (float*)d_out;

    dim3 grid(BATCH / MTILE);   // 32 workgroups, one per 16-row batch tile
    dim3 block(256);            // 8 wave32 waves, one 16x16 output tile each
    rnn_fused_kernel<<<grid, block, 0, stream>>>(xs, W_ih, W_hh, b_ih, b_hh,
                                                 W_out, b_out, out);
}